// MHAWithCache_34952443855343
// MI455X (gfx1250) — compile-verified
//
#include <hip/hip_runtime.h>
#include <hip/hip_bf16.h>

// MHA decode step (T=1) for MI455X / gfx1250, wave32.
// Memory-bound workload (~4.6 GB traffic @ 23.3 TB/s => ~195 us floor).
// fp32 WMMA (v_wmma_f32_16x16x4_f32) for all 4 projections; the cache
// concatenation is fused into the flash-decode attention kernel so
// k_cache/v_cache are read exactly once and k_cat/v_cat written once.

typedef __attribute__((ext_vector_type(2))) float v2f;
typedef __attribute__((ext_vector_type(8))) float v8f;

static constexpr int DM = 4096;   // model dim
static constexpr int NB = 8;      // batch
static constexpr int NH = 32;     // heads
static constexpr int HD = 128;    // head dim
static constexpr int TC = 8192;   // cache length
static constexpr int TK = 8193;   // cache length + 1

// ---------------------------------------------------------------------------
// Skinny GEMM: Y[0..7, :] = A[8, 4096] @ W[4096, 4096]
// One 32-column tile per block: two 16x16 accumulators share each A fetch.
// A rows 8..15 are *replicated* copies of rows 0..7 (lc & 7): output rows
// depend only on their own A row, and rows 8..15 of D are discarded at the
// store, so no zero-padding predicate (and no EXEC divergence) is needed.
// 4 waves split K (1024 each); partials reduced through LDS.
// cacheLayout==1 remaps columns into the [B,H,TK,D] append-slot layout.
// ---------------------------------------------------------------------------
__global__ __launch_bounds__(128) void gemm8_wmma_f32(
    const float* __restrict__ A, const float* __restrict__ W,
    float* __restrict__ dst, int cacheLayout, long long rowStride)
{
    const int tid  = threadIdx.x;
    const int wave = tid >> 5;
    const int lane = tid & 31;
    const int n0   = blockIdx.x << 5;   // 32-col tile
    const int half = lane >> 4;         // 0 -> K+0/1, 1 -> K+2/3 (ISA layout)
    const int lc   = lane & 15;         // A: row m; B/C: col n

    v8f accA = {};                      // cols n0 + lc
    v8f accB = {};                      // cols n0 + 16 + lc
    const float* arow = A + (size_t)(lc & 7) * DM;  // replicate rows, no mask

    const int kBegin = wave * (DM / 4); // K split across the 4 waves
    const int kEnd   = kBegin + (DM / 4);
    for (int k0 = kBegin; k0 < kEnd; k0 += 4) {
        const int ka = k0 + half * 2;
        const float2 av = *(const float2*)(arow + ka);   // unconditional b64
        v2f a;                          // A 16x4: v0 = K(+0|+2), v1 = K(+1|+3)
        a.x = av.x;
        a.y = av.y;
        const float* wp = W + (size_t)ka * DM + n0 + lc;
        v2f b0;                         // B 4x16, sub-tile 0
        b0.x = wp[0];
        b0.y = wp[DM];
        v2f b1;                         // B 4x16, sub-tile 1
        b1.x = wp[16];
        b1.y = wp[DM + 16];
        accA = __builtin_amdgcn_wmma_f32_16x16x4_f32(
            false, a, false, b0, (short)0, accA, false, false);
        accB = __builtin_amdgcn_wmma_f32_16x16x4_f32(
            false, a, false, b1, (short)0, accB, false, false);
    }

    // Cross-wave K reduction. Only rows 0..7 (lanes 0..15) carry real data.
    __shared__ float lds[4][8][33];
    if (lane < 16) {
#pragma unroll
        for (int r = 0; r < 8; ++r) {
            lds[wave][r][lc]      = accA[r];
            lds[wave][r][lc + 16] = accB[r];
        }
    }
    __syncthreads();

    if (tid < 128) {
        const int r = tid >> 4;         // batch row 0..7
#pragma unroll
        for (int cc = 0; cc < 2; ++cc) {
            const int c = (tid & 15) + cc * 16;   // col within 32-wide tile
            float s = lds[0][r][c] + lds[1][r][c] + lds[2][r][c] + lds[3][r][c];
            const int j = n0 + c;                 // global column 0..4095
            const long long coloff = cacheLayout
                ? (long long)(j >> 7) * (TK * HD) + (j & 127)  // head remap
                : (long long)j;
            dst[(long long)r * rowStride + coloff] = s;
        }
    }
}

// ---------------------------------------------------------------------------
// Flash-decode attention with fused cache concatenation.
// One block (8 wave32) per (b,h) plane. Scores pass streams k_cache rows,
// writing them into k_cat on the fly; V pass streams v_cache rows, writing
// v_cat on the fly. Row t=8192 (new token) is read back from the cat buffer
// where the projection GEMM already deposited it.
// ---------------------------------------------------------------------------
__global__ __launch_bounds__(256) void attn_decode_fused(
    const float* __restrict__ qbuf,                       // [8, 4096]
    const float* __restrict__ kcache,                     // [8,32,8192,128]
    const float* __restrict__ vcache,
    float* __restrict__ kcat,                             // [8,32,8193,128]
    float* __restrict__ vcat,
    float* __restrict__ attn_out)                         // [8, 4096]
{
    const int bh   = blockIdx.x;        // 0..255
    const int b    = bh >> 5;
    const int h    = bh & 31;
    const int tid  = threadIdx.x;
    const int wave = tid >> 5;
    const int lane = tid & 31;

    __shared__ float  sc[TK];           // 32 KB of scores (LDS: 320 KB/WGP)
    __shared__ float  red[256];
    __shared__ float4 part[8][32];
    __shared__ float  s_inv;

    const float4 q4 =
        ((const float4*)(qbuf + (size_t)b * DM + (size_t)h * HD))[lane];
    const float scale = 0.0883883476483184f; // 1/sqrt(128)

    const float* ksrc = kcache + (size_t)bh * TC * HD;
    float*       kdst = kcat   + (size_t)bh * TK * HD;

    // ---- pass 1: scores (+ fused k_cache -> k_cat copy) ----
    for (int t = wave; t < TK; t += 8) {
        float4 k4;
        if (t < TC) {
            k4 = *(const float4*)(ksrc + (size_t)t * HD + lane * 4);
            *(float4*)(kdst + (size_t)t * HD + lane * 4) = k4;   // concat
        } else {
            k4 = *(const float4*)(kdst + (size_t)t * HD + lane * 4); // new row
        }
        float s = q4.x * k4.x + q4.y * k4.y + q4.z * k4.z + q4.w * k4.w;
#pragma unroll
        for (int off = 16; off > 0; off >>= 1) s += __shfl_xor(s, off, 32);
        if (lane == 0) sc[t] = s * scale;
    }
    __syncthreads();

    // ---- softmax over 8193 scores (block reductions) ----
    float m = -__builtin_inff();
    for (int i = tid; i < TK; i += 256) m = fmaxf(m, sc[i]);
    red[tid] = m;
    __syncthreads();
    for (int s2 = 128; s2 > 0; s2 >>= 1) {
        if (tid < s2) red[tid] = fmaxf(red[tid], red[tid + s2]);
        __syncthreads();
    }
    m = red[0];
    __syncthreads();

    float lsum = 0.0f;
    for (int i = tid; i < TK; i += 256) {
        const float e = __expf(sc[i] - m);
        sc[i] = e;
        lsum += e;
    }
    red[tid] = lsum;
    __syncthreads();
    for (int s2 = 128; s2 > 0; s2 >>= 1) {
        if (tid < s2) red[tid] += red[tid + s2];
        __syncthreads();
    }
    if (tid == 0) s_inv = 1.0f / red[0];
    __syncthreads();

    // ---- pass 2: out = attn @ V (+ fused v_cache -> v_cat copy) ----
    const float* vsrc = vcache + (size_t)bh * TC * HD;
    float*       vdst = vcat   + (size_t)bh * TK * HD;
    const int    d4   = lane * 4;
    float4 acc = {0.f, 0.f, 0.f, 0.f};
    const int t0 = wave * 1025;                 // 8 segments cover 8193
    const int t1 = (t0 + 1025 < TK) ? (t0 + 1025) : TK;
    for (int t = t0; t < t1; ++t) {
        float4 v4;
        if (t < TC) {
            v4 = *(const float4*)(vsrc + (size_t)t * HD + d4);
            *(float4*)(vdst + (size_t)t * HD + d4) = v4;         // concat
        } else {
            v4 = *(const float4*)(vdst + (size_t)t * HD + d4);   // new row
        }
        const float w = sc[t];
        acc.x += w * v4.x;
        acc.y += w * v4.y;
        acc.z += w * v4.z;
        acc.w += w * v4.w;
    }
    part[wave][lane] = acc;
    __syncthreads();

    if (tid < 128) {
        const float* pf = (const float*)part;   // [8][128]
        float s = 0.0f;
#pragma unroll
        for (int g = 0; g < 8; ++g) s += pf[g * 128 + tid];
        attn_out[(size_t)b * DM + (size_t)h * HD + tid] = s * s_inv;
    }
}

// ---------------------------------------------------------------------------
extern "C" void kernel_launch(void* const* d_in, const int* in_sizes, int n_in,
                              void* d_out, int out_size, void* d_ws, size_t ws_size,
                              hipStream_t stream)
{
    (void)in_sizes; (void)n_in; (void)out_size; (void)ws_size;
    const float* x  = (const float*)d_in[0];
    const float* Wq = (const float*)d_in[1];
    const float* Wk = (const float*)d_in[2];
    const float* Wv = (const float*)d_in[3];
    const float* Wo = (const float*)d_in[4];
    const float* kc = (const float*)d_in[5];
    const float* vc = (const float*)d_in[6];

    // d_out layout: out[8*4096] | k_cat[8*32*8193*128] | v_cat[same]
    float* out  = (float*)d_out;
    float* kcat = out  + (size_t)NB * DM;
    float* vcat = kcat + (size_t)NB * NH * TK * HD;

    // scratch: q[8*4096] | attn_out[8*4096]
    float* qbuf = (float*)d_ws;
    float* abuf = qbuf + (size_t)NB * DM;

    const long long catStride = (long long)NH * TK * HD; // batch stride in cat

    // QKV projections; k,v rows land directly in their cat[t=8192] slots.
    gemm8_wmma_f32<<<dim3(DM / 32), 128, 0, stream>>>(x, Wq, qbuf, 0, (long long)DM);
    gemm8_wmma_f32<<<dim3(DM / 32), 128, 0, stream>>>(x, Wk, kcat + (size_t)TC * HD, 1, catStride);
    gemm8_wmma_f32<<<dim3(DM / 32), 128, 0, stream>>>(x, Wv, vcat + (size_t)TC * HD, 1, catStride);

    // Attention + fused cache concatenation (1 block per (b,h)).
    attn_decode_fused<<<dim3(NB * NH), 256, 0, stream>>>(qbuf, kc, vc, kcat, vcat, abuf);

    // Output projection.
    gemm8_wmma_f32<<<dim3(DM / 32), 128, 0, stream>>>(abuf, Wo, out, 0, (long long)DM);
}